// DeorphaNN_86964497809482
// MI455X (gfx1250) — compile-verified
//
#include <hip/hip_runtime.h>
#include <hip/hip_bf16.h>
#include <math.h>

#define N_NODES 20000
#define N_EDGES 200000
#define F_IN    128
#define LDS_A_STRIDE (F_IN + 4)   // 132 floats: 16B-aligned rows, conflict-free A reads
#define H_HEADS 10
#define C_HID   64
#define HC      640     // H_HEADS * C_HID
#define G_GRAPHS 64
#define NEG_SLOPE 0.2f
#define BN_EPS  1e-5f

typedef __attribute__((ext_vector_type(2))) float v2f;
typedef __attribute__((ext_vector_type(8))) float v8f;

// float atomic max via sign-aware integer punning (works with amax init = -inf)
__device__ __forceinline__ void atomic_max_f32(float* addr, float v) {
    if (v >= 0.0f) atomicMax((int*)addr, __float_as_int(v));
    else           atomicMin((unsigned int*)addr, __float_as_uint(v));
}

// ---------------------------------------------------------------------------
// 1) Init: zero accumulators, amax = -inf, fold BN into scale/shift
// ---------------------------------------------------------------------------
__global__ void k_init(float* __restrict__ nodeacc, float* __restrict__ amax,
                       float* __restrict__ denom, float* __restrict__ gsum,
                       float* __restrict__ gcnt, float* __restrict__ bnscale,
                       float* __restrict__ bnshift,
                       const float* __restrict__ bn_w, const float* __restrict__ bn_b,
                       const float* __restrict__ bn_mean, const float* __restrict__ bn_var) {
    int i = blockIdx.x * blockDim.x + threadIdx.x;
    if (i < N_NODES * C_HID)   nodeacc[i] = 0.0f;
    if (i < N_NODES * H_HEADS) { amax[i] = -INFINITY; denom[i] = 0.0f; }
    if (i < G_GRAPHS * C_HID)  gsum[i] = 0.0f;
    if (i < G_GRAPHS)          gcnt[i] = 0.0f;
    if (i < F_IN) {
        float s = bn_w[i] * rsqrtf(bn_var[i] + BN_EPS);
        bnscale[i] = s;
        bnshift[i] = bn_b[i] - bn_mean[i] * s;
    }
}

// ---------------------------------------------------------------------------
// 2) Fused dual node GEMM: Yl = BN(X)@W_l + b_l, Yr = BN(X)@W_r + b_r.
//    One BN(X) A-fragment feeds two v_wmma_f32_16x16x4_f32 per K-step.
// ---------------------------------------------------------------------------
__global__ void __launch_bounds__(256)
k_gemm_node2(const float* __restrict__ X,
             const float* __restrict__ Wl, const float* __restrict__ bl,
             const float* __restrict__ Wr, const float* __restrict__ br,
             const float* __restrict__ bnscale, const float* __restrict__ bnshift,
             float* __restrict__ Yl, float* __restrict__ Yr) {
    const int lane  = threadIdx.x & 31;
    const int wave  = threadIdx.x >> 5;
    const int m0    = blockIdx.x * 16;
    const int n0    = (blockIdx.y * 8 + wave) * 16;
    const int arow  = m0 + (lane & 15);
    const int bcol  = n0 + (lane & 15);
    const int khalf = (lane >> 4) << 1;   // lanes 16-31 hold K+2, K+3

    v8f accl = {}, accr = {};
    for (int k = 0; k < F_IN; k += 4) {
        const int kk = k + khalf;
        v2f a;
        a.x = X[arow * F_IN + kk    ] * bnscale[kk    ] + bnshift[kk    ];
        a.y = X[arow * F_IN + kk + 1] * bnscale[kk + 1] + bnshift[kk + 1];
        v2f bL, bR;
        bL.x = Wl[(kk    ) * HC + bcol];
        bL.y = Wl[(kk + 1) * HC + bcol];
        bR.x = Wr[(kk    ) * HC + bcol];
        bR.y = Wr[(kk + 1) * HC + bcol];
        accl = __builtin_amdgcn_wmma_f32_16x16x4_f32(false, a, false, bL,
                                                     (short)0, accl, false, false);
        accr = __builtin_amdgcn_wmma_f32_16x16x4_f32(false, a, false, bR,
                                                     (short)0, accr, false, false);
    }
    const float bvl   = bl[bcol];
    const float bvr   = br[bcol];
    const int   rbase = m0 + ((lane >> 4) << 3);  // lanes 16-31 hold rows M+8..M+15
#pragma unroll
    for (int r = 0; r < 8; ++r) {
        Yl[(rbase + r) * HC + bcol] = accl[r] + bvl;
        Yr[(rbase + r) * HC + bcol] = accr[r] + bvr;
    }
}

// ---------------------------------------------------------------------------
// 3) Fused edge kernel: stage the 16x128 edge_attr tile in LDS once
//    (coalesced float4, padded rows -> conflict-free ds_load_b64 A-frags),
//    WMMA e = tile @ W_e into LDS, then attention logits + segment max.
// ---------------------------------------------------------------------------
__global__ void __launch_bounds__(256)
k_edge(const float* __restrict__ EA, const float* __restrict__ We,
       const float* __restrict__ xl, const float* __restrict__ xr,
       const float* __restrict__ att, const int* __restrict__ edge_index,
       float* __restrict__ alpha, float* __restrict__ amax) {
    __shared__ float a_lds[16][LDS_A_STRIDE];  // 8.25 KB: edge_attr tile
    __shared__ float e_lds[16][HC];            // 40 KB:   e = EA @ W_e tile
    const int tid = threadIdx.x;
    const int e0  = blockIdx.x * 16;

    // -- cooperative tile stage: thread t -> row t>>4, 8 consecutive floats --
    {
        const int r  = tid >> 4;
        const int jj = (tid & 15) * 8;
        const float4* s4 = (const float4*)(EA + (size_t)(e0 + r) * F_IN + jj);
        float4 p0 = s4[0];
        float4 p1 = s4[1];
        *(float4*)&a_lds[r][jj]     = p0;
        *(float4*)&a_lds[r][jj + 4] = p1;
    }
    // prefetch next block's edge_attr tile (global_prefetch_b8) while we compute
    if (e0 + 16 < N_EDGES)
        __builtin_prefetch(EA + (size_t)(e0 + 16) * F_IN + (tid << 3), 0, 0);
    __syncthreads();

    const int lane  = tid & 31;
    const int wave  = tid >> 5;
    const int ar    = lane & 15;
    const int khalf = (lane >> 4) << 1;

    // -- GEMM phase: each wave produces 5 of the 40 N-tiles, A from LDS --
    for (int t = 0; t < 5; ++t) {
        const int n0   = (wave * 5 + t) * 16;
        const int bcol = n0 + (lane & 15);
        v8f acc = {};
        for (int k = 0; k < F_IN; k += 4) {
            const int kk = k + khalf;
            v2f a = *(const v2f*)&a_lds[ar][kk];       // ds_load_b64, bank-clean
            v2f b;
            b.x = We[(kk    ) * HC + bcol];
            b.y = We[(kk + 1) * HC + bcol];
            acc = __builtin_amdgcn_wmma_f32_16x16x4_f32(false, a, false, b,
                                                        (short)0, acc, false, false);
        }
        const int rbase = (lane >> 4) << 3;
#pragma unroll
        for (int r = 0; r < 8; ++r) e_lds[rbase + r][bcol] = acc[r];
    }
    __syncthreads();

    // -- Attention-logit phase: 16 threads per edge, shfl-reduce width 16 --
    const int i   = tid >> 4;       // edge within tile
    const int j   = tid & 15;       // worker lane
    const int e   = e0 + i;
    const int src = edge_index[e];
    const int dst = edge_index[N_EDGES + e];
    for (int h = 0; h < H_HEADS; ++h) {
        float s = 0.0f;
#pragma unroll
        for (int cc = 0; cc < 4; ++cc) {
            const int c = h * C_HID + j + cc * 16;
            float m = e_lds[i][c] + xl[(size_t)src * HC + c] + xr[(size_t)dst * HC + c];
            m = (m >= 0.0f) ? m : NEG_SLOPE * m;
            s += m * att[c];
        }
        s += __shfl_xor(s, 8, 16);
        s += __shfl_xor(s, 4, 16);
        s += __shfl_xor(s, 2, 16);
        s += __shfl_xor(s, 1, 16);
        if (j == 0) {
            alpha[e * H_HEADS + h] = s;
            atomic_max_f32(&amax[dst * H_HEADS + h], s);
        }
    }
}

// ---------------------------------------------------------------------------
// 4) Softmax numerator + denominator accumulation
// ---------------------------------------------------------------------------
__global__ void k_softmax(const int* __restrict__ edge_index,
                          const float* __restrict__ amax,
                          float* __restrict__ alpha, float* __restrict__ denom) {
    int i = blockIdx.x * blockDim.x + threadIdx.x;
    if (i >= N_EDGES * H_HEADS) return;
    int e = i / H_HEADS, h = i - e * H_HEADS;
    int dst = edge_index[N_EDGES + e];
    float ex = __expf(alpha[i] - amax[dst * H_HEADS + h]);
    alpha[i] = ex;
    atomicAdd(&denom[dst * H_HEADS + h], ex);
}

// ---------------------------------------------------------------------------
// 5) Weighted aggregation: nodeacc[dst,c] += sum_h alpha * xl[src,h,c]
// ---------------------------------------------------------------------------
__global__ void __launch_bounds__(256)
k_aggregate(const int* __restrict__ edge_index, const float* __restrict__ alpha,
            const float* __restrict__ denom, const float* __restrict__ xl,
            float* __restrict__ nodeacc) {
    const int tid = threadIdx.x;
    const int e   = blockIdx.x * 4 + (tid >> 6);
    const int c   = tid & 63;
    const int src = edge_index[e];
    const int dst = edge_index[N_EDGES + e];
    float acc = 0.0f;
#pragma unroll
    for (int h = 0; h < H_HEADS; ++h) {
        float w = alpha[e * H_HEADS + h] / (denom[dst * H_HEADS + h] + 1e-16f);
        acc += w * xl[(size_t)src * HC + h * C_HID + c];
    }
    atomicAdd(&nodeacc[dst * C_HID + c], acc);
}

// ---------------------------------------------------------------------------
// 6) Head mean + bias + ReLU + graph mean-pool accumulation
// ---------------------------------------------------------------------------
__global__ void k_node_pool(const float* __restrict__ nodeacc,
                            const float* __restrict__ bias,
                            const int* __restrict__ batch,
                            float* __restrict__ gsum, float* __restrict__ gcnt) {
    int i = blockIdx.x * blockDim.x + threadIdx.x;
    if (i >= N_NODES * C_HID) return;
    int n = i >> 6, c = i & 63;
    float v = fmaxf(nodeacc[i] * (1.0f / H_HEADS) + bias[c], 0.0f);
    int g = batch[n];
    atomicAdd(&gsum[g * C_HID + c], v);
    if (c == 0) atomicAdd(&gcnt[g], 1.0f);
}

// ---------------------------------------------------------------------------
// 7) Final linear: out[G,2] = (gsum/cnt) @ W_lin + b_lin
// ---------------------------------------------------------------------------
__global__ void k_final(const float* __restrict__ gsum, const float* __restrict__ gcnt,
                        const float* __restrict__ Wlin, const float* __restrict__ blin,
                        float* __restrict__ out) {
    int t = threadIdx.x;
    if (t >= G_GRAPHS * 2) return;
    int g = t >> 1, j = t & 1;
    float inv = 1.0f / fmaxf(gcnt[g], 1.0f);
    float s = 0.0f;
    for (int c = 0; c < C_HID; ++c)
        s += gsum[g * C_HID + c] * inv * Wlin[c * 2 + j];
    out[t] = s + blin[j];
}

// ---------------------------------------------------------------------------
extern "C" void kernel_launch(void* const* d_in, const int* in_sizes, int n_in,
                              void* d_out, int out_size, void* d_ws, size_t ws_size,
                              hipStream_t stream) {
    const float* x         = (const float*)d_in[0];
    const float* edge_attr = (const float*)d_in[1];
    const float* bn_w      = (const float*)d_in[2];
    const float* bn_b      = (const float*)d_in[3];
    const float* bn_mean   = (const float*)d_in[4];
    const float* bn_var    = (const float*)d_in[5];
    const float* W_l       = (const float*)d_in[6];
    const float* b_l       = (const float*)d_in[7];
    const float* W_r       = (const float*)d_in[8];
    const float* b_r       = (const float*)d_in[9];
    const float* W_e       = (const float*)d_in[10];
    const float* att       = (const float*)d_in[11];
    const float* bias      = (const float*)d_in[12];
    const float* W_lin     = (const float*)d_in[13];
    const float* b_lin     = (const float*)d_in[14];
    const int*   edge_index= (const int*)d_in[15];
    const int*   batch     = (const int*)d_in[16];
    float*       out       = (float*)d_out;

    // workspace carve-out (~117 MB of fp32)
    float* ws      = (float*)d_ws;
    float* x_l     = ws; ws += (size_t)N_NODES * HC;
    float* x_r     = ws; ws += (size_t)N_NODES * HC;
    float* alpha   = ws; ws += (size_t)N_EDGES * H_HEADS;
    float* amax    = ws; ws += (size_t)N_NODES * H_HEADS;
    float* denom   = ws; ws += (size_t)N_NODES * H_HEADS;
    float* nodeacc = ws; ws += (size_t)N_NODES * C_HID;
    float* gsum    = ws; ws += G_GRAPHS * C_HID;
    float* gcnt    = ws; ws += G_GRAPHS;
    float* bnscale = ws; ws += F_IN;
    float* bnshift = ws; ws += F_IN;

    k_init<<<(N_NODES * C_HID + 255) / 256, 256, 0, stream>>>(
        nodeacc, amax, denom, gsum, gcnt, bnscale, bnshift,
        bn_w, bn_b, bn_mean, bn_var);

    dim3 ggrid(N_NODES / 16, HC / (16 * 8));   // (1250, 5)
    k_gemm_node2<<<ggrid, 256, 0, stream>>>(x, W_l, b_l, W_r, b_r,
                                            bnscale, bnshift, x_l, x_r);

    k_edge<<<N_EDGES / 16, 256, 0, stream>>>(edge_attr, W_e, x_l, x_r, att,
                                             edge_index, alpha, amax);

    k_softmax<<<(N_EDGES * H_HEADS + 255) / 256, 256, 0, stream>>>(
        edge_index, amax, alpha, denom);

    k_aggregate<<<N_EDGES / 4, 256, 0, stream>>>(edge_index, alpha, denom, x_l, nodeacc);

    k_node_pool<<<(N_NODES * C_HID + 255) / 256, 256, 0, stream>>>(
        nodeacc, bias, batch, gsum, gcnt);

    k_final<<<1, 128, 0, stream>>>(gsum, gcnt, W_lin, b_lin, out);
}